// SemanticRenderer_70205535421052
// MI455X (gfx1250) — compile-verified
//
#include <hip/hip_runtime.h>

typedef __attribute__((ext_vector_type(2))) float v2f;
typedef __attribute__((ext_vector_type(8))) float v8f;

#define NCLS 32

__global__ void SemanticRenderer_zero_kernel(float* __restrict__ out, int n) {
  int i = blockIdx.x * blockDim.x + threadIdx.x;
  if (i < n) out[i] = 0.0f;
}

__device__ __forceinline__ int rl_i(int v, int l) {
  return __builtin_amdgcn_readlane(v, l);
}
__device__ __forceinline__ float rl_f(float v, int l) {
  return __int_as_float(__builtin_amdgcn_readlane(__float_as_int(v), l));
}

__device__ __forceinline__ void flush_tile(float* __restrict__ out,
                                           const v8f& d_lo, const v8f& d_hi,
                                           int rbase, int num_rays,
                                           int row, int hi) {
  if (rbase + 15 < num_rays) {           // scalar fast path: whole tile in range
#pragma unroll
    for (int v = 0; v < 8; ++v) {
      int r = rbase + v + (hi ? 8 : 0);  // D row = v (+8 for lanes 16-31)
      atomicAdd(&out[r * NCLS + row],      d_lo[v]);
      atomicAdd(&out[r * NCLS + row + 16], d_hi[v]);
    }
  } else {
#pragma unroll
    for (int v = 0; v < 8; ++v) {
      int r = rbase + v + (hi ? 8 : 0);
      if (r < num_rays) {
        atomicAdd(&out[r * NCLS + row],      d_lo[v]);
        atomicAdd(&out[r * NCLS + row + 16], d_hi[v]);
      }
    }
  }
}

__launch_bounds__(256)
__global__ void SemanticRenderer_wmma_kernel(const float* __restrict__ sem,
                                             const float* __restrict__ wgt,
                                             const int* __restrict__ ridx,
                                             float* __restrict__ out,
                                             int n_samples, int num_rays,
                                             int samples_per_wave) {
  const int lane = threadIdx.x & 31;
  // threadIdx.x>>5 is wave-uniform but divergence analysis can't see that;
  // readfirstlane pins the whole chunk bookkeeping into SGPRs.
  const int wave = __builtin_amdgcn_readfirstlane(
      blockIdx.x * (blockDim.x >> 5) + (threadIdx.x >> 5));

  int base = wave * samples_per_wave;             // SGPR
  if (base >= n_samples) return;                  // scalar branch
  int wend = base + samples_per_wave;             // SGPR
  if (wend > n_samples) wend = n_samples;

  const int row = lane & 15;                      // N column (class within half)
  const int hi  = lane >> 4;                      // 0: K=0,1 half / 1: K=2,3 half

  const v8f vzero = {};
  v8f d_lo = vzero;                               // classes 0..15 tile
  v8f d_hi = vzero;                               // classes 16..31 tile
  int rbase = __builtin_amdgcn_readfirstlane(ridx[base]);  // SGPR window base

  for (int s0 = base; s0 < wend; s0 += 32) {      // scalar loop counter
    int si  = s0 + lane;
    int sic = (si < n_samples - 1) ? si : (n_samples - 1);
    float w_raw = wgt[sic];                       // always in-bounds: no branch
    int   r_l   = ridx[sic];                      // clamped, stays sorted
    float w_l   = (si < wend) ? w_raw : 0.0f;     // cndmask, not branch

#pragma unroll
    for (int g = 0; g < 8; ++g) {
      const int k0 = g * 4;
      const int r0 = rl_i(r_l, k0 + 0);           // SGPRs
      const int r1 = rl_i(r_l, k0 + 1);
      const int r2 = rl_i(r_l, k0 + 2);
      const int r3 = rl_i(r_l, k0 + 3);

      if (r3 - rbase >= 16) {                     // scalar: window advances
        flush_tile(out, d_lo, d_hi, rbase, num_rays, row, hi);
        d_lo = vzero; d_hi = vzero;
        rbase = r0;
      }

      const float w0 = rl_f(w_l, k0 + 0);
      const float w1 = rl_f(w_l, k0 + 1);
      const float w2 = rl_f(w_l, k0 + 2);
      const float w3 = rl_f(w_l, k0 + 3);

      if (r3 - rbase >= 16) {                     // rare: empty-ray jump in group
        const int   rr[4] = {r0, r1, r2, r3};
        const float ww[4] = {w0, w1, w2, w3};
#pragma unroll
        for (int k = 0; k < 4; ++k) {
          if (rr[k] - rbase >= 16) {              // scalar; exact fallback
            int srow = s0 + k0 + k;
            if (srow < wend) {
              float v = sem[(size_t)srow * NCLS + lane];
              atomicAdd(&out[rr[k] * NCLS + lane], ww[k] * v);
            }
            // A-side auto-masked: (rr[k]-rbase) > 15 never matches a row index
          }
        }
      }

      // A (16x4 f32): lanes 0-15 hold K=0,1 for row M=lane; lanes 16-31 hold K=2,3
      int   rka = hi ? r2 : r0;
      int   rkb = hi ? r3 : r1;
      float wka = hi ? w2 : w0;
      float wkb = hi ? w3 : w1;
      v2f a;
      a.x = (rka - rbase == row) ? wka : 0.0f;
      a.y = (rkb - rbase == row) ? wkb : 0.0f;

      // B (4x16 f32): lane = column; VGPR0/1 = K rows (0,1) low half, (2,3) high half
      int srow0 = s0 + k0 + (hi ? 2 : 0);
      if (srow0 > n_samples - 2) srow0 = n_samples - 2;  // clamp; masked by zero A
      const float* p0 = sem + (size_t)srow0 * NCLS;
      v2f b_lo, b_hi;
      b_lo.x = p0[row];           b_lo.y = p0[NCLS + row];
      b_hi.x = p0[row + 16];      b_hi.y = p0[NCLS + row + 16];

      // D += A x B  (exact f32 accumulate on the matrix pipe)
      d_lo = __builtin_amdgcn_wmma_f32_16x16x4_f32(false, a, false, b_lo,
                                                   (short)0, d_lo, false, false);
      d_hi = __builtin_amdgcn_wmma_f32_16x16x4_f32(false, a, false, b_hi,
                                                   (short)0, d_hi, false, false);
    }
  }
  flush_tile(out, d_lo, d_hi, rbase, num_rays, row, hi);
}

extern "C" void kernel_launch(void* const* d_in, const int* in_sizes, int n_in,
                              void* d_out, int out_size, void* d_ws, size_t ws_size,
                              hipStream_t stream) {
  const float* sem  = (const float*)d_in[0];   // [N, 32] f32
  const float* wgt  = (const float*)d_in[1];   // [N] f32
  const int*   ridx = (const int*)d_in[2];     // [N] int (sorted)
  float* out = (float*)d_out;                  // [num_rays, 32] f32

  const int n        = in_sizes[1];            // N_SAMPLES
  const int num_rays = out_size / NCLS;

  // d_out is poisoned by the harness: zero it first (same stream -> ordered)
  SemanticRenderer_zero_kernel<<<(out_size + 255) / 256, 256, 0, stream>>>(out, out_size);

  const int spw    = 512;                      // samples per wave (multiple of 32)
  const int waves  = (n + spw - 1) / spw;      // 8192 waves for N=4M
  const int blocks = (waves + 7) / 8;          // 8 waves (256 thr) per block
  SemanticRenderer_wmma_kernel<<<blocks, 256, 0, stream>>>(
      sem, wgt, ridx, out, n, num_rays, spw);
}